// HAN_24661702214218
// MI455X (gfx1250) — compile-verified
//
#include <hip/hip_runtime.h>
#include <cmath>

typedef __attribute__((ext_vector_type(2))) float v2f;
typedef __attribute__((ext_vector_type(8))) float v8f;

#define HID 128
#define NH 8
#define HD 16

static __device__ __forceinline__ unsigned fkey(float x) {
    unsigned b = __float_as_uint(x);
    return b ^ ((b >> 31) ? 0xFFFFFFFFu : 0x80000000u);
}
static __device__ __forceinline__ float funkey(unsigned k) {
    unsigned b = (k >> 31) ? (k ^ 0x80000000u) : ~k;
    return __uint_as_float(b);
}

__global__ void han_zero(float* __restrict__ p, size_t n) {
    size_t i = (size_t)blockIdx.x * blockDim.x + threadIdx.x;
    if (i < n) p[i] = 0.0f;
}

// P[N,128] = X[N,128] @ W[128,128] + bias.
// mode 0: store P. mode 1: tanh(P) column-summed into colsum[128] (P unused).
// One wave per 16-row strip; 8 f32 accumulators cover all 128 columns.
__global__ void han_gemm128_wmma(const float* __restrict__ X, const float* __restrict__ W,
                                 const float* __restrict__ bias, float* __restrict__ P,
                                 float* __restrict__ colsum, int N, int mode)
{
    int lane = threadIdx.x & 31;
    int wave = blockIdx.x * (blockDim.x >> 5) + (threadIdx.x >> 5);
    int m0 = wave * 16;
    if (m0 >= N) return;              // wave-uniform: EXEC stays all-ones for WMMA
    int hi = lane >> 4;               // half-wave select
    int lo = lane & 15;

    v8f acc[8] = {};

    for (int k0 = 0; k0 < HID; k0 += 4) {
        int ka = k0 + 2 * hi;
        const float* xr = X + (size_t)(m0 + lo) * HID + ka;
        v2f a;
        a.x = xr[0];                  // A[M=lo][ka]
        a.y = xr[1];                  // A[M=lo][ka+1]
        #pragma unroll
        for (int t = 0; t < 8; ++t) {
            v2f b;
            b.x = W[(size_t)ka * HID + t * 16 + lo];        // B[ka][n]
            b.y = W[(size_t)(ka + 1) * HID + t * 16 + lo];  // B[ka+1][n]
            acc[t] = __builtin_amdgcn_wmma_f32_16x16x4_f32(
                false, a, false, b, (short)0, acc[t], false, false);
        }
    }

    if (mode == 0) {
        #pragma unroll
        for (int t = 0; t < 8; ++t) {
            float bv = bias[t * 16 + lo];
            #pragma unroll
            for (int r = 0; r < 8; ++r)
                P[(size_t)(m0 + r + 8 * hi) * HID + t * 16 + lo] = acc[t][r] + bv;
        }
    } else {
        #pragma unroll
        for (int t = 0; t < 8; ++t) {
            float bv = bias[t * 16 + lo];
            float s = 0.0f;
            #pragma unroll
            for (int r = 0; r < 8; ++r) s += tanhf(acc[t][r] + bv);
            s += __shfl_xor(s, 16, 32);          // combine M=r and M=r+8 halves
            if (hi == 0) atomicAdd(&colsum[t * 16 + lo], s);
        }
    }
}

// dots[n][h] = sum_d P[n][h*16+d] * att[h][d]
__global__ void han_head_dots(const float* __restrict__ P, const float* __restrict__ att,
                              float* __restrict__ dots, int N)
{
    int i = blockIdx.x * blockDim.x + threadIdx.x;
    if (i >= N * NH) return;
    int n = i >> 3, h = i & 7;
    const float* pr = P + (size_t)n * HID + h * HD;
    const float* ar = att + h * HD;
    float s = 0.0f;
    #pragma unroll
    for (int d = 0; d < HD; ++d) s += pr[d] * ar[d];
    dots[i] = s;
}

// logit = leaky_relu(as[src]+ad[dst]); store; atomic ordered-uint max per (dst,head)
__global__ void han_edge_logits_max(const float* __restrict__ asv, const float* __restrict__ adv,
                                    const int* __restrict__ src, const int* __restrict__ dst,
                                    float* __restrict__ elog, unsigned* __restrict__ maxb, int EH)
{
    int i = blockIdx.x * blockDim.x + threadIdx.x;
    if (i >= EH) return;
    int e = i >> 3, h = i & 7;
    float a = asv[(size_t)src[e] * NH + h] + adv[(size_t)dst[e] * NH + h];
    a = (a > 0.0f) ? a : 0.2f * a;
    elog[i] = a;
    atomicMax(&maxb[(size_t)dst[e] * NH + h], fkey(a));
}

// e = exp(logit - max[dst]); overwrite elog; atomicAdd into segment sums
__global__ void han_edge_exp_sum(const int* __restrict__ dst, float* __restrict__ elog,
                                 const unsigned* __restrict__ maxb, float* __restrict__ ssum, int EH)
{
    int i = blockIdx.x * blockDim.x + threadIdx.x;
    if (i >= EH) return;
    int e = i >> 3, h = i & 7;
    int d = dst[e];
    float m = funkey(maxb[(size_t)d * NH + h]);
    if (!(fabsf(m) < 3.0e38f)) m = 0.0f;     // non-finite -> 0 (matches reference)
    float ev = expf(elog[i] - m);
    elog[i] = ev;
    atomicAdd(&ssum[(size_t)d * NH + h], ev);
}

// out[dst][c] += xs[src][c] * (e / s[dst][head(c)]) ; one thread per (edge, channel)
__global__ void han_edge_scatter(const float* __restrict__ xs, const float* __restrict__ elog,
                                 const float* __restrict__ ssum,
                                 const int* __restrict__ src, const int* __restrict__ dst,
                                 float* __restrict__ out, int EC)
{
    int i = blockIdx.x * blockDim.x + threadIdx.x;
    if (i >= EC) return;
    int e = i >> 7, c = i & 127, h = c >> 4;
    int d = dst[e], s = src[e];
    float alpha = elog[(size_t)e * NH + h] / ssum[(size_t)d * NH + h];
    atomicAdd(&out[(size_t)d * HID + c], xs[(size_t)s * HID + c] * alpha);
}

__global__ void han_relu2(float* __restrict__ a, float* __restrict__ b, size_t n) {
    size_t i = (size_t)blockIdx.x * blockDim.x + threadIdx.x;
    if (i < n) { a[i] = fmaxf(a[i], 0.0f); b[i] = fmaxf(b[i], 0.0f); }
}

// score_r = (colsum_r/N) . q ; beta = softmax over 2 relations
__global__ void han_beta(const float* __restrict__ colsum, const float* __restrict__ q,
                         float invN, float* __restrict__ beta)
{
    __shared__ float sh0[128], sh1[128];
    int t = threadIdx.x;
    float qv = q[t];
    sh0[t] = colsum[t] * invN * qv;
    sh1[t] = colsum[HID + t] * invN * qv;
    __syncthreads();
    for (int off = 64; off > 0; off >>= 1) {
        if (t < off) { sh0[t] += sh0[t + off]; sh1[t] += sh1[t + off]; }
        __syncthreads();
    }
    if (t == 0) {
        float s0 = sh0[0], s1 = sh1[0];
        float m = fmaxf(s0, s1);
        float e0 = expf(s0 - m), e1 = expf(s1 - m);
        float inv = 1.0f / (e0 + e1);
        beta[0] = e0 * inv; beta[1] = e1 * inv;
    }
}

// out[n,0:2] = (beta0*o0[n] + beta1*o1[n]) @ w_out + b_out ; one wave per node
__global__ void han_final(const float* __restrict__ o0, const float* __restrict__ o1,
                          const float* __restrict__ beta, const float* __restrict__ wout,
                          const float* __restrict__ bout, float* __restrict__ out, int N)
{
    int lane = threadIdx.x & 31;
    int wave = (blockIdx.x * blockDim.x + threadIdx.x) >> 5;
    if (wave >= N) return;
    float b0 = beta[0], b1 = beta[1];
    const float4 f0 = ((const float4*)(o0 + (size_t)wave * HID))[lane];
    const float4 f1 = ((const float4*)(o1 + (size_t)wave * HID))[lane];
    int c0 = lane * 4;
    float s0 = 0.0f, s1 = 0.0f, v;
    v = b0 * f0.x + b1 * f1.x; s0 += v * wout[(c0 + 0) * 2]; s1 += v * wout[(c0 + 0) * 2 + 1];
    v = b0 * f0.y + b1 * f1.y; s0 += v * wout[(c0 + 1) * 2]; s1 += v * wout[(c0 + 1) * 2 + 1];
    v = b0 * f0.z + b1 * f1.z; s0 += v * wout[(c0 + 2) * 2]; s1 += v * wout[(c0 + 2) * 2 + 1];
    v = b0 * f0.w + b1 * f1.w; s0 += v * wout[(c0 + 3) * 2]; s1 += v * wout[(c0 + 3) * 2 + 1];
    #pragma unroll
    for (int off = 16; off > 0; off >>= 1) {
        s0 += __shfl_xor(s0, off, 32);
        s1 += __shfl_xor(s1, off, 32);
    }
    if (lane == 0) {
        out[(size_t)wave * 2 + 0] = s0 + bout[0];
        out[(size_t)wave * 2 + 1] = s1 + bout[1];
    }
}

static inline int ceil_div(long long a, long long b) { return (int)((a + b - 1) / b); }

extern "C" void kernel_launch(void* const* d_in, const int* in_sizes, int n_in,
                              void* d_out, int out_size, void* d_ws, size_t ws_size,
                              hipStream_t stream)
{
    const float* x_user = (const float*)d_in[0];
    const float* x_item = (const float*)d_in[1];
    const float* w_user = (const float*)d_in[2];
    const float* b_user = (const float*)d_in[3];
    const float* w_item = (const float*)d_in[4];
    const float* b_item = (const float*)d_in[5];
    const float* att_src_iu = (const float*)d_in[6];
    const float* att_dst_iu = (const float*)d_in[7];
    const float* att_src_uu = (const float*)d_in[8];
    const float* att_dst_uu = (const float*)d_in[9];
    const float* wk    = (const float*)d_in[10];
    const float* bk    = (const float*)d_in[11];
    const float* q     = (const float*)d_in[12];
    const float* w_out = (const float*)d_in[13];
    const float* b_out = (const float*)d_in[14];
    const int* e_iu_src = (const int*)d_in[15];
    const int* e_iu_dst = (const int*)d_in[16];
    const int* e_uu_src = (const int*)d_in[17];
    const int* e_uu_dst = (const int*)d_in[18];

    const int N_U = in_sizes[0] / HID;
    const int N_I = in_sizes[1] / HID;
    const int E   = in_sizes[15];
    float* out = (float*)d_out;

    // workspace carve-up
    float* ws = (float*)d_ws;
    size_t off = 0;
    float* xu    = ws + off; off += (size_t)N_U * HID;
    float* xi    = ws + off; off += (size_t)N_I * HID;
    float* o_iu  = ws + off; off += (size_t)N_U * HID;
    float* o_uu  = ws + off; off += (size_t)N_U * HID;
    float* ad_iu = ws + off; off += (size_t)N_U * NH;
    float* as_uu = ws + off; off += (size_t)N_U * NH;
    float* ad_uu = ws + off; off += (size_t)N_U * NH;
    float* as_iu = ws + off; off += (size_t)N_I * NH;
    float* elog  = ws + off; off += (size_t)E * NH;
    unsigned* maxb = (unsigned*)(ws + off); off += (size_t)N_U * NH;
    float* ssum  = ws + off; off += (size_t)N_U * NH;   // contiguous after maxb
    float* colsum = ws + off; off += 2 * HID;
    float* beta  = ws + off; off += 2;

    const int B = 256;
    const size_t oN = (size_t)N_U * HID;

    // zero accumulators (o_iu/o_uu contiguous; colsum)
    han_zero<<<ceil_div(2 * (long long)oN, B), B, 0, stream>>>(o_iu, 2 * oN);
    han_zero<<<1, B, 0, stream>>>(colsum, 2 * HID);

    // projections (WMMA fp32 GEMMs)
    int gb_u = ceil_div((long long)N_U / 16, 8);
    int gb_i = ceil_div((long long)N_I / 16, 8);
    han_gemm128_wmma<<<gb_u, B, 0, stream>>>(x_user, w_user, b_user, xu, nullptr, N_U, 0);
    han_gemm128_wmma<<<gb_i, B, 0, stream>>>(x_item, w_item, b_item, xi, nullptr, N_I, 0);

    // per-(node,head) attention dots
    han_head_dots<<<ceil_div((long long)N_I * NH, B), B, 0, stream>>>(xi, att_src_iu, as_iu, N_I);
    han_head_dots<<<ceil_div((long long)N_U * NH, B), B, 0, stream>>>(xu, att_dst_iu, ad_iu, N_U);
    han_head_dots<<<ceil_div((long long)N_U * NH, B), B, 0, stream>>>(xu, att_src_uu, as_uu, N_U);
    han_head_dots<<<ceil_div((long long)N_U * NH, B), B, 0, stream>>>(xu, att_dst_uu, ad_uu, N_U);

    const int EH = E * NH;
    const int EC = E * HID;
    const int segN = 2 * N_U * NH;   // maxb + ssum contiguous

    // relation item->user
    han_zero<<<ceil_div(segN, B), B, 0, stream>>>((float*)maxb, (size_t)segN);
    han_edge_logits_max<<<ceil_div(EH, B), B, 0, stream>>>(as_iu, ad_iu, e_iu_src, e_iu_dst, elog, maxb, EH);
    han_edge_exp_sum<<<ceil_div(EH, B), B, 0, stream>>>(e_iu_dst, elog, maxb, ssum, EH);
    han_edge_scatter<<<ceil_div(EC, B), B, 0, stream>>>(xi, elog, ssum, e_iu_src, e_iu_dst, o_iu, EC);

    // relation user->user
    han_zero<<<ceil_div(segN, B), B, 0, stream>>>((float*)maxb, (size_t)segN);
    han_edge_logits_max<<<ceil_div(EH, B), B, 0, stream>>>(as_uu, ad_uu, e_uu_src, e_uu_dst, elog, maxb, EH);
    han_edge_exp_sum<<<ceil_div(EH, B), B, 0, stream>>>(e_uu_dst, elog, maxb, ssum, EH);
    han_edge_scatter<<<ceil_div(EC, B), B, 0, stream>>>(xu, elog, ssum, e_uu_src, e_uu_dst, o_uu, EC);

    // relu on relation outputs
    han_relu2<<<ceil_div((long long)oN, B), B, 0, stream>>>(o_iu, o_uu, oN);

    // semantic attention: tanh(o@wk+bk) column-means via fused WMMA GEMM
    han_gemm128_wmma<<<gb_u, B, 0, stream>>>(o_iu, wk, bk, nullptr, colsum, N_U, 1);
    han_gemm128_wmma<<<gb_u, B, 0, stream>>>(o_uu, wk, bk, nullptr, colsum + HID, N_U, 1);
    han_beta<<<1, 128, 0, stream>>>(colsum, q, 1.0f / (float)N_U, beta);

    // fused output projection
    han_final<<<ceil_div((long long)N_U * 32, B), B, 0, stream>>>(o_iu, o_uu, beta, w_out, b_out, out, N_U);
}